// DUQHead_63367947485245
// MI455X (gfx1250) — compile-verified
//
#include <hip/hip_runtime.h>
#include <stdint.h>

typedef float v2f __attribute__((ext_vector_type(2)));
typedef float v4f __attribute__((ext_vector_type(4)));
typedef float v8f __attribute__((ext_vector_type(8)));
typedef unsigned int u32x4 __attribute__((ext_vector_type(4)));
typedef int i32x4 __attribute__((ext_vector_type(4)));
typedef int i32x8 __attribute__((ext_vector_type(8)));

// Problem constants: features [8,512,128,128], weights [16,64,512]
#define NFEAT 512        // K
#define HW    16384      // 128*128
#define NCLS  64         // classes
#define NEMB  16         // embedding dim e
#define WAVES 4          // one 16-pixel M-tile per wave
#define GAMMA_OVER_E (-3.125f)   // (-1/(2*0.1^2)) / 16

// B half-slice in LDS, written by TDM with pad_enable:
// 32 class-rows x 512 floats, +2 dwords inserted every 256 dwords streamed
// -> row stride 516 floats, and +2 extra offset for f >= 256 within a row.
#define BROW  516
#define BHALF (32 * BROW)        // 16512 floats = 64.5 KB

// ---- Tensor Data Mover: load one 32x512 f32 weight slice into LDS (padded).
// D# packing per CDNA5 ISA ch.8 (group0 128b, group1 256b; groups 2/3 unused-2D).
__device__ __forceinline__ void tdm_load_b(uint32_t lds_addr, const float* gsrc) {
    const uint64_t ga = (uint64_t)(uintptr_t)gsrc;
    u32x4 g0;
    g0.x = 1u;                                   // count=1 (valid user descriptor)
    g0.y = lds_addr;                             // LDS byte address
    g0.z = (uint32_t)ga;                         // global_addr[31:0]
    g0.w = (uint32_t)((ga >> 32) & 0x1FFFFFFull) // global_addr[56:32]
         | (2u << 30);                           // type=2 ("image")
    i32x8 g1;
    g1[0] = (2 << 16)      // data_size = 4 bytes
          | (1 << 20)      // pad_enable
          | (7 << 22)      // pad_interval = 256 dwords
          | (1 << 25);     // pad_amount   = 2 dwords
    g1[1] = (int)(512u << 16);            // tensor_dim0[15:0] = 512  (bits 63:48)
    g1[2] = (int)(32u << 16);             // tensor_dim1[15:0] = 32   (bits 95:80)
    g1[3] = (int)(512u << 16);            // tile_dim0 = 512          (bits 127:112)
    g1[4] = (int)(32u);                   // tile_dim1 = 32, tile_dim2 = 0
    g1[5] = (int)(512);                   // tensor_dim0_stride = 512
    g1[6] = 0;
    g1[7] = 0;
    i32x4 gz; gz[0] = 0; gz[1] = 0; gz[2] = 0; gz[3] = 0;   // groups 2/3 unused (2-D)
#if __has_include(<hip/amd_detail/amd_gfx1250_TDM.h>)
    i32x8 gz8; for (int i = 0; i < 8; ++i) gz8[i] = 0;
    __builtin_amdgcn_tensor_load_to_lds(g0, g1, gz, gz, gz8, 0);
#else
    __builtin_amdgcn_tensor_load_to_lds(g0, g1, gz, gz, 0);
#endif
}

__global__ __launch_bounds__(128) void duq_rbf_kernel(
    const float* __restrict__ feat,   // [8,512,128,128]
    const float* __restrict__ wts,    // [16,64,512]
    const float* __restrict__ mbuf,   // [16,64]
    const float* __restrict__ nbuf,   // [64]
    float* __restrict__ out)          // [8,64,128,128]
{
    __shared__ float Alds[WAVES][8192];   // 128 KB: per-wave A tile, WMMA frag layout
    __shared__ float Bbuf[2][BHALF];      // 129 KB: TDM ping-pong weight slices

    const int tid  = threadIdx.x;
    const int wid  = tid >> 5;
    const int lane = tid & 31;
    const int n    = lane & 15;     // column within a 16-wide c-tile
    const int h    = lane >> 4;

    const int tileId = blockIdx.x * WAVES + wid;   // 16-pixel M-tile id
    const int pbase  = tileId * 16;
    const int b      = pbase >> 14;                // / HW
    const int hw0    = pbase & (HW - 1);
    const float* fbase = feat + (size_t)b * NFEAT * HW + hw0;

    // Kick off TDM for stage 0 (e=0, classes 0..31) before A staging: overlap.
    if (wid == 0) {
        tdm_load_b((uint32_t)(uintptr_t)&Bbuf[0][0], wts);
    }

    // ---- Stage A: 16 pixels x 512 features -> WMMA A-fragment layout in LDS.
    //  frag(chunk q, lane L=(m,h)) = {A[m][4q+2h], A[m][4q+2h+1]}
    //  float index = q2*128 + h*64 + m*4 + (q&1)*2 + (f&1)   (q2 = q>>1)
    float* Aw = Alds[wid];
    for (int it = 0; it < 64; ++it) {
        const int t = it * 32 + lane;     // f = t>>2, j = t&3 (pixels m=4j..4j+3)
        const int f = t >> 2;
        const int j = t & 3;
        v4f val = *(const v4f*)(fbase + (size_t)f * HW + 4 * j);
        const int q2 = f >> 3;
        const int qo = (f >> 2) & 1;
        const int hh = (f >> 1) & 1;
        const int v  = f & 1;
        const int base = q2 * 128 + hh * 64 + (4 * j) * 4 + qo * 2 + v;
        Aw[base]      = val.x;
        Aw[base + 4]  = val.y;
        Aw[base + 8]  = val.z;
        Aw[base + 12] = val.w;
    }

    float invN[4];
    #pragma unroll
    for (int ct = 0; ct < 4; ++ct) invN[ct] = 1.0f / nbuf[ct * 16 + n];

    v8f sq[4];
    #pragma unroll
    for (int ct = 0; ct < 4; ++ct)
        #pragma unroll
        for (int i = 0; i < 8; ++i) sq[ct][i] = 0.0f;

    const float* ap = Aw + lane * 4;
    const float* bbase[2] = { &Bbuf[0][n * BROW + 2 * h],
                              &Bbuf[1][n * BROW + 2 * h] };

    // One stage = (e, class-half cp). cp is always a literal at the call sites,
    // so sq[] indexing and the ping-pong buffer selection are fully static.
    auto do_stage = [&](int e, int cp, bool issueNext) {
        if (wid == 0) __builtin_amdgcn_s_wait_tensorcnt(0);  // TDM(this) complete
        __syncthreads();   // publish Bbuf[cp]; all waves done reading Bbuf[cp^1]

        if (wid == 0 && issueNext) {   // overlap next TDM with this compute
            const int e1  = (cp == 0) ? e : e + 1;
            const int cp1 = cp ^ 1;
            tdm_load_b((uint32_t)(uintptr_t)&Bbuf[cp ^ 1][0],
                       wts + ((size_t)e1 * NCLS + cp1 * 32) * NFEAT);
        }

        // ---- GEMM slice: emb[16 pix x 32 cls] for (e, cp). K = 512.
        v8f acc0, acc1;
        #pragma unroll
        for (int i = 0; i < 8; ++i) { acc0[i] = 0.0f; acc1[i] = 0.0f; }

        const float* bp = bbase[cp];

        #pragma unroll 2
        for (int q2 = 0; q2 < 32; ++q2) {           // f in [0,256): no pad offset
            v4f a   = *(const v4f*)(ap + q2 * 128);
            v2f b00 = *(const v2f*)(bp + q2 * 8);
            v2f b01 = *(const v2f*)(bp + q2 * 8 + 4);
            v2f b10 = *(const v2f*)(bp + 16 * BROW + q2 * 8);
            v2f b11 = *(const v2f*)(bp + 16 * BROW + q2 * 8 + 4);
            acc0 = __builtin_amdgcn_wmma_f32_16x16x4_f32(false, (v2f)a.lo, false, b00, (short)0, acc0, false, false);
            acc1 = __builtin_amdgcn_wmma_f32_16x16x4_f32(false, (v2f)a.lo, false, b10, (short)0, acc1, false, false);
            acc0 = __builtin_amdgcn_wmma_f32_16x16x4_f32(false, (v2f)a.hi, false, b01, (short)0, acc0, false, false);
            acc1 = __builtin_amdgcn_wmma_f32_16x16x4_f32(false, (v2f)a.hi, false, b11, (short)0, acc1, false, false);
        }
        #pragma unroll 2
        for (int q2 = 32; q2 < 64; ++q2) {          // f in [256,512): +2 dword pad
            v4f a   = *(const v4f*)(ap + q2 * 128);
            v2f b00 = *(const v2f*)(bp + q2 * 8 + 2);
            v2f b01 = *(const v2f*)(bp + q2 * 8 + 6);
            v2f b10 = *(const v2f*)(bp + 16 * BROW + q2 * 8 + 2);
            v2f b11 = *(const v2f*)(bp + 16 * BROW + q2 * 8 + 6);
            acc0 = __builtin_amdgcn_wmma_f32_16x16x4_f32(false, (v2f)a.lo, false, b00, (short)0, acc0, false, false);
            acc1 = __builtin_amdgcn_wmma_f32_16x16x4_f32(false, (v2f)a.lo, false, b10, (short)0, acc1, false, false);
            acc0 = __builtin_amdgcn_wmma_f32_16x16x4_f32(false, (v2f)a.hi, false, b01, (short)0, acc0, false, false);
            acc1 = __builtin_amdgcn_wmma_f32_16x16x4_f32(false, (v2f)a.hi, false, b11, (short)0, acc1, false, false);
        }

        // ---- Epilogue: sq += (emb - m[e,c]/N[c])^2  (static sq indices)
        const int ct0 = cp * 2, ct1 = cp * 2 + 1;
        const float cm0 = mbuf[e * NCLS + ct0 * 16 + n] * invN[ct0];
        const float cm1 = mbuf[e * NCLS + ct1 * 16 + n] * invN[ct1];
        #pragma unroll
        for (int i = 0; i < 8; ++i) {
            const float d0 = acc0[i] - cm0;
            const float d1 = acc1[i] - cm1;
            sq[ct0][i] += d0 * d0;
            sq[ct1][i] += d1 * d1;
        }
    };

    for (int e = 0; e < NEMB; ++e) {
        do_stage(e, 0, true);            // classes  0..31, buffer 0
        do_stage(e, 1, e < NEMB - 1);    // classes 32..63, buffer 1
    }

    // ---- Final: out[b,c,hw0+pm] = exp(gamma/16 * sum_e diff^2)
    // D layout: VGPR i -> pixel i + 8*h, lane&15 -> class column.
    float* obase = out + (size_t)b * NCLS * HW + hw0 + h * 8;
    #pragma unroll
    for (int ct = 0; ct < 4; ++ct) {
        const int c = ct * 16 + n;
        #pragma unroll
        for (int i = 0; i < 8; ++i) {
            obase[(size_t)c * HW + i] = __expf(sq[ct][i] * GAMMA_OVER_E);
        }
    }
}

extern "C" void kernel_launch(void* const* d_in, const int* in_sizes, int n_in,
                              void* d_out, int out_size, void* d_ws, size_t ws_size,
                              hipStream_t stream) {
    const float* feat = (const float*)d_in[0];   // [8,512,128,128]
    const float* wts  = (const float*)d_in[1];   // [16,64,512]
    const float* m    = (const float*)d_in[2];   // [16,64]
    const float* N    = (const float*)d_in[3];   // [64]
    float* out = (float*)d_out;                  // [8,64,128,128] f32

    // 131072 pixels / (4 waves * 16 pixels) = 2048 blocks of 128 threads
    dim3 grid(2048), block(128);
    hipLaunchKernelGGL(duq_rbf_kernel, grid, block, 0, stream,
                       feat, wts, m, N, out);
}